// FeatureField_16286515987045
// MI455X (gfx1250) — compile-verified
//
#include <hip/hip_runtime.h>
#include <stdint.h>

// Hashgrid feature interpolation (single-level InstantNGP encoding).
// N = 4194304 points, table = 2^19 x float2 (4 MB, L2-resident on MI455X).
// Memory-bound: ~84 MB HBM streaming -> ~3.6us floor @ 23.3 TB/s.
// Dominant cost: 32M random 8B gathers hitting L2 -> maximize MLP:
// per thread we issue all 32 gathers (4 points x 8 corners) before consuming.
// CDNA5 paths: global_load_async_to_lds_b128 (ASYNCcnt) x-tile staging,
// s_wait_asynccnt, NT hints on the streaming loads/stores. No matmul
// structure in this workload -> WMMA is inapplicable.

#define LOG2_HASHMAP_SIZE 19
#define HASH_MASK ((1u << LOG2_HASHMAP_SIZE) - 1u)
#define RES_F 128.0f
#define P1 2654435761u
#define P2 805459861u

#define BLOCK 256
#define TILE 1024              // points per block
#define ITEMS (TILE / BLOCK)   // 4 points per thread

typedef float v2f __attribute__((ext_vector_type(2)));
typedef float v4f __attribute__((ext_vector_type(4)));
typedef int   v4i __attribute__((ext_vector_type(4)));

typedef __attribute__((address_space(1))) v4i* as1_v4i_ptr;
typedef __attribute__((address_space(3))) v4i* as3_v4i_ptr;

#if defined(__AMDGCN__) && __has_builtin(__builtin_amdgcn_global_load_async_to_lds_b128)
#define HAVE_ASYNC_LDS 1
#else
#define HAVE_ASYNC_LDS 0
#endif

__device__ __forceinline__ void wait_async_zero() {
#if __has_builtin(__builtin_amdgcn_s_wait_asynccnt)
    __builtin_amdgcn_s_wait_asynccnt(0);
#elif defined(__AMDGCN__)
    asm volatile("s_wait_asynccnt 0" ::: "memory");
#endif
}

// Phase 1 result for one point: 8 table indices + interpolation weights.
struct PointCtx {
    uint32_t h[8];
    float dx, dy, dz;
};

__device__ __forceinline__ PointCtx hash_point(float x0, float x1, float x2) {
    PointCtx c;
    float xs = x0 * RES_F, ys = x1 * RES_F, zs = x2 * RES_F;
    float fxf = floorf(xs), fyf = floorf(ys), fzf = floorf(zs);
    float cxf = ceilf(xs),  cyf = ceilf(ys),  czf = ceilf(zs);
    uint32_t fx = (uint32_t)(int)fxf, fy = (uint32_t)(int)fyf, fz = (uint32_t)(int)fzf;
    uint32_t cx = (uint32_t)(int)cxf, cy = (uint32_t)(int)cyf, cz = (uint32_t)(int)czf;

    uint32_t hfy = fy * P1, hcy = cy * P1;   // CSE: each y/z product used 4x
    uint32_t hfz = fz * P2, hcz = cz * P2;
    uint32_t a00 = fx ^ hfy, a10 = cx ^ hfy, a01 = fx ^ hcy, a11 = cx ^ hcy;

    c.h[0] = (a00 ^ hfz) & HASH_MASK;  // 000
    c.h[1] = (a10 ^ hfz) & HASH_MASK;  // 100
    c.h[2] = (a01 ^ hfz) & HASH_MASK;  // 010
    c.h[3] = (a00 ^ hcz) & HASH_MASK;  // 001
    c.h[4] = (a11 ^ hfz) & HASH_MASK;  // 110
    c.h[5] = (a10 ^ hcz) & HASH_MASK;  // 101
    c.h[6] = (a01 ^ hcz) & HASH_MASK;  // 011
    c.h[7] = (a11 ^ hcz) & HASH_MASK;  // 111
    c.dx = xs - fxf; c.dy = ys - fyf; c.dz = zs - fzf;
    return c;
}

// Phase 3: trilinear interpolation, corner order 000,100,010,001,110,101,011,111.
__device__ __forceinline__ v2f lerp8(const v2f v[8], float dx, float dy, float dz) {
    float ox = 1.0f - dx, oy = 1.0f - dy, oz = 1.0f - dz;
    v2f c00 = v[0] * ox + v[1] * dx;   // 000/100
    v2f c01 = v[3] * ox + v[5] * dx;   // 001/101
    v2f c10 = v[2] * ox + v[4] * dx;   // 010/110
    v2f c11 = v[6] * ox + v[7] * dx;   // 011/111
    v2f c0  = c00 * oy + c10 * dy;
    v2f c1  = c01 * oy + c11 * dy;
    return c0 * oz + c1 * dz;
}

__device__ __forceinline__ v2f interp_point(float x0, float x1, float x2,
                                            const v2f* __restrict__ ht) {
    PointCtx c = hash_point(x0, x1, x2);
    v2f v[8];
#pragma unroll
    for (int j = 0; j < 8; ++j) v[j] = ht[c.h[j]];
    return lerp8(v, c.dx, c.dy, c.dz);
}

__global__ __launch_bounds__(BLOCK) void hashgrid_interp_kernel(
    const float* __restrict__ xg,      // [npts,3]
    const v2f*   __restrict__ ht,      // [2^19] float2
    v2f*         __restrict__ outg,    // [npts] float2
    int npts) {
    const int t = threadIdx.x;
    const long long tile_base = (long long)blockIdx.x * TILE;
    const bool full_tile = (tile_base + TILE) <= (long long)npts;

#if HAVE_ASYNC_LDS
    __shared__ v4f sx4[TILE * 3 / 4];  // 12 KB x-tile staged in LDS
    if (full_tile) {
        const v4f* gsrc = (const v4f*)(xg + tile_base * 3);
#pragma unroll
        for (int k = 0; k < 3; ++k) {
            int j = t + k * BLOCK;  // 256 lanes x 16B x 3 = 12288 B, fully coalesced
            __builtin_amdgcn_global_load_async_to_lds_b128(
                (as1_v4i_ptr)(uintptr_t)(gsrc + j),
                (as3_v4i_ptr)(uint32_t)(uintptr_t)(sx4 + j),
                0, 0);
        }
        wait_async_zero();   // own wave's async copies done
        __syncthreads();     // all waves' copies visible
    }
    const float* sx = (const float*)sx4;
#endif

    if (full_tile) {
        // ---- Phase 1: hashes + weights for all ITEMS points ----
        PointCtx ctx[ITEMS];
#pragma unroll
        for (int k = 0; k < ITEMS; ++k) {
            const int p = t + k * BLOCK;  // consecutive lanes -> consecutive points
            float x0, x1, x2;
#if HAVE_ASYNC_LDS
            // stride-3-dword LDS reads: gcd(3,64)=1 -> bank-conflict-free
            x0 = sx[3 * p + 0];
            x1 = sx[3 * p + 1];
            x2 = sx[3 * p + 2];
#else
            const float* xp = xg + (tile_base + p) * 3;
            __builtin_prefetch(xg + (tile_base + TILE) * 3, 0, 1);  // global_prefetch_b8
            x0 = __builtin_nontemporal_load(xp + 0);
            x1 = __builtin_nontemporal_load(xp + 1);
            x2 = __builtin_nontemporal_load(xp + 2);
#endif
            ctx[k] = hash_point(x0, x1, x2);
        }

        // ---- Phase 2: issue all 32 gathers back-to-back (max MLP) ----
        v2f val[ITEMS][8];
#pragma unroll
        for (int k = 0; k < ITEMS; ++k)
#pragma unroll
            for (int j = 0; j < 8; ++j)
                val[k][j] = ht[ctx[k].h[j]];

        // ---- Phase 3: interpolate + NT stream out ----
#pragma unroll
        for (int k = 0; k < ITEMS; ++k) {
            const int p = t + k * BLOCK;
            v2f o = lerp8(val[k], ctx[k].dx, ctx[k].dy, ctx[k].dz);
            __builtin_nontemporal_store(o, outg + tile_base + p);
        }
    } else {
        // Tail tile (not hit for N = 2^22, kept for generality)
        for (int k = 0; k < ITEMS; ++k) {
            const long long i = tile_base + t + k * BLOCK;
            if (i < (long long)npts) {
                const float* xp = xg + i * 3;
                v2f o = interp_point(xp[0], xp[1], xp[2], ht);
                __builtin_nontemporal_store(o, outg + i);
            }
        }
    }
}

extern "C" void kernel_launch(void* const* d_in, const int* in_sizes, int n_in,
                              void* d_out, int out_size, void* d_ws, size_t ws_size,
                              hipStream_t stream) {
    const float* x  = (const float*)d_in[0];   // [N,3] float32
    const v2f*   ht = (const v2f*)d_in[1];     // [2^19,2] float32
    v2f* out = (v2f*)d_out;                    // [N,2] float32
    const int npts = in_sizes[0] / 3;
    const int blocks = (npts + TILE - 1) / TILE;
    hashgrid_interp_kernel<<<blocks, BLOCK, 0, stream>>>(x, ht, out, npts);
}